// Trainer_68169720922260
// MI455X (gfx1250) — compile-verified
//
#include <hip/hip_runtime.h>
#include <hip/hip_bf16.h>
#include <stdint.h>

#define T_STEPS 1024
#define ROWS_PER_BLOCK 4          // 4 wave32s per block, one row per wave
#define LMBDA 0.95f

// ---- CDNA5 async global<->LDS helpers (inline asm; tracked by ASYNCcnt) ----
__device__ __forceinline__ unsigned lds_off_of(const void* p) {
  // generic pointer to LDS: low 32 bits are the wave-relative LDS byte offset
  return (unsigned)(uintptr_t)p;
}
__device__ __forceinline__ void async_ld_b128(unsigned lds, const void* g) {
  asm volatile("global_load_async_to_lds_b128 %0, %1, off" :: "v"(lds), "v"(g) : "memory");
}
__device__ __forceinline__ void async_ld_b32(unsigned lds, const void* g) {
  asm volatile("global_load_async_to_lds_b32 %0, %1, off" :: "v"(lds), "v"(g) : "memory");
}
__device__ __forceinline__ void async_st_b128(void* g, unsigned lds) {
  asm volatile("global_store_async_from_lds_b128 %0, %1, off" :: "v"(g), "v"(lds) : "memory");
}
__device__ __forceinline__ void async_st_b32(void* g, unsigned lds) {
  asm volatile("global_store_async_from_lds_b32 %0, %1, off" :: "v"(g), "v"(lds) : "memory");
}
__device__ __forceinline__ void wait_async0() {
  asm volatile("s_wait_asynccnt 0x0" ::: "memory");
}
__device__ __forceinline__ void wait_ds0() {
  asm volatile("s_wait_dscnt 0x0" ::: "memory");
}

__global__ __launch_bounds__(ROWS_PER_BLOCK * 32)
void vtrace_scan_kernel(const float* __restrict__ gammas,
                        const float* __restrict__ vf_x,
                        const float* __restrict__ rewards,
                        const float* __restrict__ action_lls,
                        const float* __restrict__ orig_lls,
                        const int*   __restrict__ terminated,
                        const int*   __restrict__ episode_lengths,
                        float*       __restrict__ out,
                        int n_rows) {
  __shared__ __align__(16) float smem[ROWS_PER_BLOCK * 3 * T_STEPS];

  const int lane = threadIdx.x & 31;
  const int w    = threadIdx.x >> 5;
  const int n    = blockIdx.x * ROWS_PER_BLOCK + w;
  if (n >= n_rows) return;

  float* sg = smem + w * 3 * T_STEPS;   // gammas  -> later advantages staging
  float* sr = sg + T_STEPS;             // rewards -> later v_s staging
  float* sv = sr + T_STEPS;             // vf_x[0..T-1]

  const float* gam_g = gammas   + (size_t)n * T_STEPS;
  const float* rew_g = rewards  + (size_t)n * T_STEPS;
  const float* vf_g  = vf_x     + (size_t)n * (T_STEPS + 1);
  const float* all_g = action_lls + (size_t)n * T_STEPS;
  const float* oll_g = orig_lls   + (size_t)n * T_STEPS;

  const unsigned lg = lds_off_of(sg);
  const unsigned lr = lds_off_of(sr);
  const unsigned lv = lds_off_of(sv);

  // ---- Stage twice-read inputs into LDS with async copies (coalesced 512B/wave) ----
  #pragma unroll
  for (int j = 0; j < 8; ++j) {                       // 16B-aligned rows: B128
    const unsigned o = (unsigned)(lane * 16 + j * 512);
    async_ld_b128(lg + o, (const char*)gam_g + o);
    async_ld_b128(lr + o, (const char*)rew_g + o);
  }
  #pragma unroll
  for (int j = 0; j < 32; ++j) {                      // vf row stride 4100B: only 4B-aligned -> B32
    const unsigned o = (unsigned)(lane * 4 + j * 128);
    async_ld_b32(lv + o, (const char*)vf_g + o);
  }

  // ---- Read-once inputs: direct vectorized loads (overlap with async copies) ----
  float a_l[32], o_l[32];
  {
    const float4* a4 = (const float4*)all_g + lane * 8;
    const float4* o4 = (const float4*)oll_g + lane * 8;
    #pragma unroll
    for (int j = 0; j < 8; ++j) {
      *(float4*)(a_l + 4 * j) = a4[j];
      *(float4*)(o_l + 4 * j) = o4[j];
    }
  }
  const int   last_t  = episode_lengths[n] - 1;
  const int   term    = terminated[n];
  const float vf_last = vf_g[T_STEPS];

  wait_async0();  // staged data visible in LDS (same-wave producer, no barrier needed)

  // ---- Pass 1: per-lane chunk [base, base+32): affine map  v_start = B + A * carry ----
  const int base = lane * 32;
  float rho_l[32], g_l[32];
  float A = 1.0f, B = 0.0f;
  #pragma unroll
  for (int k = 31; k >= 0; --k) {
    const int t = base + k;
    float g = sg[t];
    if (t == last_t && term) g = 0.0f;                // zero discount at terminal step
    const float iw  = __expf(a_l[k] - o_l[k]);
    const float rho = fminf(iw, 1.0f);                // RHO_MAX == C_MAX == 1
    const float gc  = LMBDA * rho * g;
    const float vfn = (t == T_STEPS - 1) ? vf_last : sv[t + 1];
    const float dV  = rho * (sr[t] + g * vfn - sv[t]);
    B = dV + gc * B;                                  // prepend f_t (outermost)
    A = gc * A;
    rho_l[k] = rho;
    g_l[k]   = g;
  }

  // ---- Wave32 suffix scan of affine maps (carry flows lane31 -> lane0) ----
  float sA = A, sB = B;
  #pragma unroll
  for (int off = 1; off < 32; off <<= 1) {
    float nA = __shfl_down(sA, (unsigned)off, 32);
    float nB = __shfl_down(sB, (unsigned)off, 32);
    const bool valid = (lane + off) < 32;
    nA = valid ? nA : 1.0f;                           // identity map past the edge
    nB = valid ? nB : 0.0f;
    sB = sB + sA * nB;                                // self ∘ neighbor
    sA = sA * nA;
  }
  float carry = __shfl_down(sB, 1u, 32);              // suffix starting at lane+1, applied to 0
  if (lane == 31) carry = 0.0f;                       // v_diff[T] = 0

  // ---- Pass 2: final recurrence + outputs, staged back into freed LDS slots ----
  float v = carry;                                    // v_diff at t = base + 32
  #pragma unroll
  for (int k = 31; k >= 0; --k) {
    const int t = base + k;
    const float rho = rho_l[k], g = g_l[k];
    const float r   = sr[t];
    const float vft = sv[t];
    const float vfn = (t == T_STEPS - 1) ? vf_last : sv[t + 1];
    const float vs_next = v + vfn;                    // v_s[t+1]
    const float adv = rho * (r * g * vs_next - vft);
    const float dV  = rho * (r + g * vfn - vft);
    v = dV + LMBDA * rho * g * v;                     // v_diff[t]
    sg[t] = adv;                                      // advantages staging
    sr[t] = v + vft;                                  // v_s staging
  }

  wait_ds0();  // LDS writes complete before async stores read LDS

  float* adv_g = out + (size_t)n * T_STEPS;
  float* vs_g  = out + (size_t)n_rows * T_STEPS + (size_t)n * (T_STEPS + 1);
  #pragma unroll
  for (int j = 0; j < 8; ++j) {
    const unsigned o = (unsigned)(lane * 16 + j * 512);
    async_st_b128((char*)adv_g + o, lg + o);
  }
  #pragma unroll
  for (int j = 0; j < 32; ++j) {                      // v_s row base only 4B-aligned -> B32
    const unsigned o = (unsigned)(lane * 4 + j * 128);
    async_st_b32((char*)vs_g + o, lr + o);
  }
  if (lane == 31) vs_g[T_STEPS] = vf_last;            // v_s[T] = vf_x[T] (v_diff[T] = 0)

  wait_async0();  // drain async stores (s_endpgm would also wait, be explicit)
}

extern "C" void kernel_launch(void* const* d_in, const int* in_sizes, int n_in,
                              void* d_out, int out_size, void* d_ws, size_t ws_size,
                              hipStream_t stream) {
  (void)n_in; (void)out_size; (void)d_ws; (void)ws_size;
  const float* gammas   = (const float*)d_in[0];
  const float* vf_x     = (const float*)d_in[1];
  const float* rewards  = (const float*)d_in[2];
  const float* act_lls  = (const float*)d_in[3];
  const float* orig_lls = (const float*)d_in[4];
  const int*   term     = (const int*)d_in[5];
  const int*   ep_len   = (const int*)d_in[6];
  float* out = (float*)d_out;

  const int n_rows = in_sizes[5];                     // N (terminated has N elements)
  const int blocks = (n_rows + ROWS_PER_BLOCK - 1) / ROWS_PER_BLOCK;
  vtrace_scan_kernel<<<blocks, ROWS_PER_BLOCK * 32, 0, stream>>>(
      gammas, vf_x, rewards, act_lls, orig_lls, term, ep_len, out, n_rows);
}